// SpGAT_13778255085709
// MI455X (gfx1250) — compile-verified
//
#include <hip/hip_runtime.h>
#include <hip/hip_bf16.h>

#define NFEAT  512
#define NCLASS 256

typedef float v2f __attribute__((ext_vector_type(2)));
typedef float v8f __attribute__((ext_vector_type(8)));

// ---------------------------------------------------------------------------
// Kernel 1: h = x @ W  via V_WMMA_F32_16X16X4_F32
// One wave -> 16 rows x 64 cols (4 N-tiles, A-frag reused across 4 B-frags).
// Block = 256 threads = 8 waves = 128 rows. gridDim.y covers NCLASS/64 = 4.
// ---------------------------------------------------------------------------
__global__ void spgat_gemm(const float* __restrict__ x,
                           const float* __restrict__ W,
                           float* __restrict__ h, int N) {
    const int lane = threadIdx.x & 31;
    const int wave = threadIdx.x >> 5;
    const int mtile = blockIdx.x * 8 + wave;
    const int mtiles = N >> 4;                 // N = 50000 -> 3125 tiles exactly
    if (mtile >= mtiles) return;
    const int m0   = mtile << 4;
    const int n0   = blockIdx.y * 64;
    const int half = lane >> 4;                // 0: K pair {0,1}; 1: K pair {2,3}
    const int sub  = lane & 15;                // M index (A) / N index (B)

    v8f acc0 = {}, acc1 = {}, acc2 = {}, acc3 = {};

    const float* xrow = x + (size_t)(m0 + sub) * NFEAT + 2 * half;
    for (int kb = 0; kb < NFEAT; kb += 4) {
        // A fragment: 16x4 f32. VGPR0 = K{0|2}, VGPR1 = K{1|3} by lane half.
        v2f afrag;
        afrag.x = xrow[kb + 0];
        afrag.y = xrow[kb + 1];

        // B fragments: 4x16 f32 per N-tile. K striped like A, N over lanes.
        const float* wk0 = W + (size_t)(kb + 2 * half + 0) * NCLASS + n0 + sub;
        const float* wk1 = W + (size_t)(kb + 2 * half + 1) * NCLASS + n0 + sub;
        v2f b0; b0.x = wk0[ 0]; b0.y = wk1[ 0];
        v2f b1; b1.x = wk0[16]; b1.y = wk1[16];
        v2f b2; b2.x = wk0[32]; b2.y = wk1[32];
        v2f b3; b3.x = wk0[48]; b3.y = wk1[48];

        acc0 = __builtin_amdgcn_wmma_f32_16x16x4_f32(false, afrag, false, b0, (short)0, acc0, false, false);
        acc1 = __builtin_amdgcn_wmma_f32_16x16x4_f32(false, afrag, false, b1, (short)0, acc1, false, false);
        acc2 = __builtin_amdgcn_wmma_f32_16x16x4_f32(false, afrag, false, b2, (short)0, acc2, false, false);
        acc3 = __builtin_amdgcn_wmma_f32_16x16x4_f32(false, afrag, false, b3, (short)0, acc3, false, false);
    }

    // C/D layout: VGPR r: lanes 0-15 -> M=r, lanes 16-31 -> M=8+r; N = sub.
    float* hout = h + n0 + sub;
    #pragma unroll
    for (int r = 0; r < 8; ++r) {
        const size_t m = (size_t)(m0 + half * 8 + r);
        hout[m * NCLASS +  0] = acc0[r];
        hout[m * NCLASS + 16] = acc1[r];
        hout[m * NCLASS + 32] = acc2[r];
        hout[m * NCLASS + 48] = acc3[r];
    }
}

// ---------------------------------------------------------------------------
// Kernel 2: s_row = h @ a[:256], s_col = h @ a[256:]. One wave32 per row.
// ---------------------------------------------------------------------------
__global__ void spgat_scores(const float* __restrict__ h,
                             const float* __restrict__ a,
                             float* __restrict__ s_row,
                             float* __restrict__ s_col, int N) {
    const int lane = threadIdx.x & 31;
    const int row  = blockIdx.x * (blockDim.x >> 5) + (threadIdx.x >> 5);
    if (row >= N) return;
    const float* hr = h + (size_t)row * NCLASS;
    float sr = 0.f, sc = 0.f;
    #pragma unroll
    for (int i = 0; i < NCLASS / 32; ++i) {
        const int f = lane + 32 * i;
        const float v = hr[f];
        sr += v * a[f];
        sc += v * a[NCLASS + f];
    }
    #pragma unroll
    for (int off = 16; off; off >>= 1) {
        sr += __shfl_xor(sr, off, 32);
        sc += __shfl_xor(sc, off, 32);
    }
    if (lane == 0) { s_row[row] = sr; s_col[row] = sc; }
}

// ---------------------------------------------------------------------------
// Kernel 3: per-edge attention weight e = exp(-leakyrelu(s_row[r]+s_col[c]))
//           and rowsum accumulation.
// ---------------------------------------------------------------------------
__global__ void spgat_edges(const int* __restrict__ ei, int E,
                            const float* __restrict__ s_row,
                            const float* __restrict__ s_col,
                            float* __restrict__ e_buf,
                            float* __restrict__ rowsum) {
    const int e = blockIdx.x * blockDim.x + threadIdx.x;
    if (e >= E) return;
    const int r = ei[e];
    const int c = ei[E + e];
    const float lg = s_row[r] + s_col[c];
    const float l  = lg > 0.f ? lg : 0.2f * lg;   // leaky_relu, slope 0.2
    const float ev = __expf(-l);
    e_buf[e] = ev;
    atomicAdd(rowsum + r, ev);
}

// ---------------------------------------------------------------------------
// Kernel 4: scatter h_prime[row] += e * h[col]. 64 threads/edge, float4 gather
// + 4 global f32 atomics per thread. This is the bandwidth-bound stage.
// ---------------------------------------------------------------------------
__global__ void spgat_scatter(const int* __restrict__ ei, int E,
                              const float* __restrict__ e_buf,
                              const float* __restrict__ h,
                              float* __restrict__ out) {
    const unsigned tid = blockIdx.x * blockDim.x + threadIdx.x;
    const int e  = tid >> 6;          // 64 threads per edge
    const int fg = tid & 63;          // 4 features per thread
    if (e >= E) return;
    const int r = ei[e];
    const int c = ei[E + e];
    const float ev = e_buf[e];
    const float4 hv = *(const float4*)(h + (size_t)c * NCLASS + fg * 4);
    float* o = out + (size_t)r * NCLASS + fg * 4;
    atomicAdd(o + 0, ev * hv.x);
    atomicAdd(o + 1, ev * hv.y);
    atomicAdd(o + 2, ev * hv.z);
    atomicAdd(o + 3, ev * hv.w);
}

// ---------------------------------------------------------------------------
// Kernel 5: out = log_softmax(elu(out / rowsum)) per row. One wave32 per row,
// 8 elements/lane, shuffle reductions for max and sum.
// ---------------------------------------------------------------------------
__global__ void spgat_finalize(float* __restrict__ out,
                               const float* __restrict__ rowsum, int N) {
    const int lane = threadIdx.x & 31;
    const int row  = blockIdx.x * (blockDim.x >> 5) + (threadIdx.x >> 5);
    if (row >= N) return;
    const float inv = 1.0f / rowsum[row];
    float* o = out + (size_t)row * NCLASS;
    float v[NCLASS / 32];
    float mx = -3.402823466e38f;
    #pragma unroll
    for (int i = 0; i < NCLASS / 32; ++i) {
        const float hp = o[lane + 32 * i] * inv;
        const float el = hp > 0.f ? hp : (__expf(hp) - 1.f);   // elu
        v[i] = el;
        mx = fmaxf(mx, el);
    }
    #pragma unroll
    for (int off = 16; off; off >>= 1) mx = fmaxf(mx, __shfl_xor(mx, off, 32));
    float s = 0.f;
    #pragma unroll
    for (int i = 0; i < NCLASS / 32; ++i) s += __expf(v[i] - mx);
    #pragma unroll
    for (int off = 16; off; off >>= 1) s += __shfl_xor(s, off, 32);
    const float lse = mx + __logf(s);
    #pragma unroll
    for (int i = 0; i < NCLASS / 32; ++i) o[lane + 32 * i] = v[i] - lse;
}

// ---------------------------------------------------------------------------
extern "C" void kernel_launch(void* const* d_in, const int* in_sizes, int n_in,
                              void* d_out, int out_size, void* d_ws, size_t ws_size,
                              hipStream_t stream) {
    const float* x  = (const float*)d_in[0];
    const float* W  = (const float*)d_in[1];
    const float* a  = (const float*)d_in[2];
    const int*   ei = (const int*)d_in[3];

    const int N = in_sizes[0] / NFEAT;     // 50000
    const int E = in_sizes[3] / 2;         // 850000

    // Workspace layout (floats): h | s_row | s_col | rowsum | e_buf
    float* h      = (float*)d_ws;
    float* s_row  = h + (size_t)N * NCLASS;
    float* s_col  = s_row + N;
    float* rowsum = s_col + N;
    float* e_buf  = rowsum + N;

    float* out = (float*)d_out;

    // Zero the atomic accumulators every call (deterministic under graph replay).
    hipMemsetAsync(out, 0, (size_t)N * NCLASS * sizeof(float), stream);
    hipMemsetAsync(rowsum, 0, (size_t)N * sizeof(float), stream);

    // 1) h = x @ W  (WMMA f32)
    {
        const int mtiles = N / 16;                     // 3125
        dim3 grid((mtiles + 7) / 8, NCLASS / 64);      // (391, 4)
        spgat_gemm<<<grid, 256, 0, stream>>>(x, W, h, N);
    }
    // 2) row/col scores
    {
        dim3 grid((N + 7) / 8);
        spgat_scores<<<grid, 256, 0, stream>>>(h, a, s_row, s_col, N);
    }
    // 3) edge weights + rowsum
    {
        dim3 grid((E + 255) / 256);
        spgat_edges<<<grid, 256, 0, stream>>>(ei, E, s_row, s_col, e_buf, rowsum);
    }
    // 4) weighted scatter (bandwidth-bound stage)
    {
        const long long threads = (long long)E * 64;
        dim3 grid((unsigned)((threads + 255) / 256));
        spgat_scatter<<<grid, 256, 0, stream>>>(ei, E, e_buf, h, out);
    }
    // 5) normalize + elu + log_softmax
    {
        dim3 grid((N + 7) / 8);
        spgat_finalize<<<grid, 256, 0, stream>>>(out, rowsum, N);
    }
}